// Model_6210522710673
// MI455X (gfx1250) — compile-verified
//
#include <hip/hip_runtime.h>
#include <hip/hip_bf16.h>
#include <math.h>

typedef __attribute__((ext_vector_type(2))) float v2f;
typedef __attribute__((ext_vector_type(8))) float v8f;
typedef __attribute__((__vector_size__(4 * sizeof(int)))) int v4i;

#define BATCH   8192
#define SEQ     512
#define PRED    720
#define NEXP    16
#define HALF_N  256
#define PI_F    3.14159265358979323846f

// ---------------- CDNA5 async global->LDS helpers ----------------
__device__ __forceinline__ void async_b128(const float* g, float* l) {
#if __has_builtin(__builtin_amdgcn_global_load_async_to_lds_b128)
    __builtin_amdgcn_global_load_async_to_lds_b128(
        (__attribute__((address_space(1))) v4i*)(unsigned long long)g,
        (__attribute__((address_space(3))) v4i*)(unsigned int)(unsigned long long)l,
        0, 0);
#else
    asm volatile("global_load_async_to_lds_b128 %0, %1, off"
                 :: "v"((unsigned int)(unsigned long long)(void*)l),
                    "v"((unsigned long long)(const void*)g)
                 : "memory");
#endif
}
__device__ __forceinline__ void wait_async9() {
#if __has_builtin(__builtin_amdgcn_s_wait_asynccnt)
    __builtin_amdgcn_s_wait_asynccnt(9);
#else
    asm volatile("s_wait_asynccnt 0x9" ::: "memory");
#endif
}
__device__ __forceinline__ void wait_async0() {
#if __has_builtin(__builtin_amdgcn_s_wait_asynccnt)
    __builtin_amdgcn_s_wait_asynccnt(0);
#else
    asm volatile("s_wait_asynccnt 0x0" ::: "memory");
#endif
}

// ---------------------------------------------------------------- stage 0
__global__ void moe_stage0_zero(int* __restrict__ counts) {
    if (threadIdx.x < NEXP) counts[threadIdx.x] = 0;
}

// ---------------------------------------------------------------- stage 1
// One 256-thread block per row: RevIN stats, xn write, 512-pt FFT periodogram,
// gating, top-2 softmax, routing-list append.
__global__ __launch_bounds__(256) void moe_stage1(
    const float* __restrict__ x, const float* __restrict__ gw, const float* __restrict__ gb,
    float* __restrict__ xn,
    int* __restrict__ e0a, int* __restrict__ e1a,
    float* __restrict__ g0a, float* __restrict__ g1a,
    float* __restrict__ mua, float* __restrict__ sda,
    int* __restrict__ counts, int* __restrict__ rowlist)
{
    __shared__ float sx[SEQ];
    __shared__ float sr[SEQ];
    __shared__ float si[SEQ];
    __shared__ float red[256];
    __shared__ float sg[256];
    __shared__ float gate[NEXP];

    const int b = blockIdx.x;
    const int t = threadIdx.x;
    const float* xb = x + (size_t)b * SEQ;

    float v0 = xb[t], v1 = xb[t + 256];
    sx[t] = v0; sx[t + 256] = v1;
    red[t] = v0 + v1;
    __syncthreads();
    for (int off = 128; off > 0; off >>= 1) {
        if (t < off) red[t] += red[t + off];
        __syncthreads();
    }
    const float mu = red[0] * (1.0f / SEQ);
    __syncthreads();

    const float d0 = v0 - mu, d1 = v1 - mu;
    red[t] = d0 * d0 + d1 * d1;
    __syncthreads();
    for (int off = 128; off > 0; off >>= 1) {
        if (t < off) red[t] += red[t + off];
        __syncthreads();
    }
    const float sd = sqrtf(red[0] * (1.0f / SEQ)) + 1e-5f;
    __syncthreads();

    // write normalized rows for the GEMM stage
    const float isd = 1.0f / sd;
    xn[(size_t)b * SEQ + t]       = d0 * isd;
    xn[(size_t)b * SEQ + t + 256] = d1 * isd;

    // bit-reversed load of mean-centered signal (real input, imag = 0)
    {
        const int i0 = t, i1 = t + 256;
        const int r0 = (int)(__brev((unsigned)i0) >> 23);
        const int r1 = (int)(__brev((unsigned)i1) >> 23);
        sr[i0] = sx[r0] - mu; si[i0] = 0.0f;
        sr[i1] = sx[r1] - mu; si[i1] = 0.0f;
    }
    __syncthreads();

    // radix-2 DIT FFT, 9 stages, 256 butterflies/stage = 1 per thread
    for (int s = 1; s <= 9; ++s) {
        const int m2 = 1 << (s - 1);
        const int jm = t & (m2 - 1);
        const int k  = ((t >> (s - 1)) << s) + jm;
        const float ang = -PI_F * (float)jm / (float)m2;
        float sn, cs;
        __sincosf(ang, &sn, &cs);
        const float vr = sr[k + m2], vi = si[k + m2];
        const float tr = cs * vr - sn * vi;
        const float ti = cs * vi + sn * vr;
        const float ur = sr[k], ui = si[k];
        sr[k] = ur + tr; si[k] = ui + ti;
        sr[k + m2] = ur - tr; si[k + m2] = ui - ti;
        __syncthreads();
    }

    // periodogram bins 0..255 (|X|^2 / N), plus sum for normalization
    const float Ik = (sr[t] * sr[t] + si[t] * si[t]) * (1.0f / SEQ);
    sx[t] = Ik;        // reuse sx[0..255] for I
    red[t] = Ik;
    __syncthreads();
    for (int off = 128; off > 0; off >>= 1) {
        if (t < off) red[t] += red[t + off];
        __syncthreads();
    }
    const float ssum = red[0];
    const float invs = (ssum == 0.0f) ? 1.0f : (1.0f / ssum);
    __syncthreads();

    // gate logits: 16 threads per expert, strided 256-dot
    const int ge = t >> 4, gj = t & 15;
    {
        const float* gwe = gw + (size_t)ge * HALF_N;
        float p = 0.0f;
        for (int kk = gj; kk < HALF_N; kk += 16) p += sx[kk] * gwe[kk];
        sg[t] = p;
    }
    __syncthreads();
    for (int off = 8; off > 0; off >>= 1) {
        if (gj < off) sg[t] += sg[t + off];
        __syncthreads();
    }
    if (gj == 0) gate[ge] = sg[ge * 16] * invs + gb[ge];
    __syncthreads();

    if (t == 0) {
        float b0 = -INFINITY, b1 = -INFINITY;
        int i0 = 0, i1 = 0;
        for (int e = 0; e < NEXP; ++e) {
            const float g = gate[e];
            if (g > b0)      { b1 = b0; i1 = i0; b0 = g; i0 = e; }
            else if (g > b1) { b1 = g; i1 = e; }
        }
        const float ex = __expf(b1 - b0);          // <= 1
        const float gg0 = 1.0f / (1.0f + ex);
        const float gg1 = ex * gg0;
        e0a[b] = i0; e1a[b] = i1;
        g0a[b] = gg0; g1a[b] = gg1;
        mua[b] = mu; sda[b] = sd;
        int p0 = atomicAdd(&counts[i0], 1);
        rowlist[i0 * BATCH + p0] = (b << 1);
        int p1 = atomicAdd(&counts[i1], 1);
        rowlist[i1 * BATCH + p1] = (b << 1) | 1;
    }
}

// ---------------------------------------------------------------- stage 2
// Gathered GEMM per expert via V_WMMA_F32_16X16X4_F32, with double-buffered
// LDS tiles filled by GLOBAL_LOAD_ASYNC_TO_LDS_B128 (ASYNCcnt pipelined).
// Block: 256 threads = 8 waves; tile = 16 gathered rows x 128 cols.
#define AS_STRIDE 68
#define BS_STRIDE 132
__global__ __launch_bounds__(256) void moe_stage2(
    const float* __restrict__ xn, const float* __restrict__ ew,
    const int* __restrict__ counts, const int* __restrict__ rowlist,
    float* __restrict__ gemm)
{
    const int cb = blockIdx.x;          // 0..5   : 128-col block
    const int rt = blockIdx.y;          // 0..511 : 16-row tile
    const int e  = blockIdx.z;          // expert
    const int cnt  = counts[e];
    const int base = rt * 16;
    if (base >= cnt) return;

    __shared__ float As[2][16 * AS_STRIDE];   // 2 x 4.3 KB
    __shared__ float Bs[2][64 * BS_STRIDE];   // 2 x 33.8 KB
    __shared__ int rowS[16], slotS[16], valS[16];

    const int t = threadIdx.x;
    if (t < 16) {
        const int idx = base + t;
        const int src = (idx < cnt) ? idx : base;      // duplicate row for padding
        const int entry = rowlist[e * BATCH + src];
        rowS[t]  = entry >> 1;
        slotS[t] = entry & 1;
        valS[t]  = (idx < cnt) ? 1 : 0;
    }
    __syncthreads();

    const float* W = ew + (size_t)e * SEQ * PRED;
    const int l = t & 31, w = t >> 5;
    const int col0 = cb * 128 + w * 16;
    const int m  = l & 15;               // A-operand row (M)
    const int kb = (l >> 4) * 2;         // K sub-offset within the 4-step
    const int nn = l & 15;               // B/D column within subtile
    const int mb = (l >> 4) * 8;         // D M-base

    // A loader mapping: 16 threads per gathered row, one B128 per thread
    const int ar = t >> 4, akq = t & 15;
    const float* arow = xn + (size_t)rowS[ar] * SEQ + akq * 4;

    // issue 9 async B128 copies (1 A + 8 B) for K-chunk kc into buffer buf
    auto issue_chunk = [&](int buf, int kc) {
        async_b128(arow + kc, &As[buf][ar * AS_STRIDE + akq * 4]);
        #pragma unroll
        for (int i = 0; i < 8; ++i) {
            const int idx = t + i * 256;
            const int kr = idx >> 5;      // 0..63
            const int cq = idx & 31;      // 0..31
            int gc = cb * 128 + cq * 4;
            if (gc >= PRED) gc = PRED - 4;   // clamp; values unused (tail waves skip)
            async_b128(W + (size_t)(kc + kr) * PRED + gc,
                       &Bs[buf][kr * BS_STRIDE + cq * 4]);
        }
    };

    issue_chunk(0, 0);
    v8f acc = {};
    for (int kc = 0; kc < SEQ; kc += 64) {
        const int buf = (kc >> 6) & 1;
        if (kc + 64 < SEQ) {
            issue_chunk(buf ^ 1, kc + 64);   // prefetch next chunk
            wait_async9();                   // oldest 9 (current chunk) complete
        } else {
            wait_async0();
        }
        __syncthreads();                     // all waves' copies visible
        if (col0 < PRED) {                   // wave-uniform; EXEC stays all-ones
            const float* Ab = &As[buf][0];
            const float* Bb = &Bs[buf][0];
            #pragma unroll
            for (int ks = 0; ks < 16; ++ks) {
                v2f a = *(const v2f*)(&Ab[m * AS_STRIDE + ks * 4 + kb]);
                v2f bvec;
                bvec.x = Bb[(ks * 4 + kb)     * BS_STRIDE + w * 16 + nn];
                bvec.y = Bb[(ks * 4 + kb + 1) * BS_STRIDE + w * 16 + nn];
                acc = __builtin_amdgcn_wmma_f32_16x16x4_f32(
                    false, a, false, bvec, (short)0, acc, false, false);
            }
        }
        __syncthreads();                     // buffer free before next overwrite
    }

    if (col0 < PRED) {
        #pragma unroll
        for (int r = 0; r < 8; ++r) {
            const int mi = mb + r;
            if (valS[mi]) {
                gemm[(size_t)slotS[mi] * BATCH * PRED +
                     (size_t)rowS[mi] * PRED + col0 + nn] = acc[r];
            }
        }
    }
}

// ---------------------------------------------------------------- stage 3
__global__ __launch_bounds__(256) void moe_stage3(
    const float* __restrict__ gemm, const float* __restrict__ eb,
    const int* __restrict__ e0a, const int* __restrict__ e1a,
    const float* __restrict__ g0a, const float* __restrict__ g1a,
    const float* __restrict__ mua, const float* __restrict__ sda,
    float* __restrict__ out)
{
    const int b = blockIdx.x, t = threadIdx.x;
    const int e0 = e0a[b], e1 = e1a[b];
    const float sd = sda[b];
    const float c0 = g0a[b] * sd, c1 = g1a[b] * sd;
    const float mu = mua[b];
    const float* G0 = gemm + (size_t)b * PRED;
    const float* G1 = gemm + (size_t)BATCH * PRED + (size_t)b * PRED;
    const float* b0 = eb + (size_t)e0 * PRED;
    const float* b1 = eb + (size_t)e1 * PRED;
    for (int p = t; p < PRED; p += 256) {
        out[(size_t)b * PRED + p] =
            c0 * (G0[p] + b0[p]) + c1 * (G1[p] + b1[p]) + mu;
    }
}

// ---------------------------------------------------------------- launch
extern "C" void kernel_launch(void* const* d_in, const int* in_sizes, int n_in,
                              void* d_out, int out_size, void* d_ws, size_t ws_size,
                              hipStream_t stream) {
    (void)in_sizes; (void)n_in; (void)out_size; (void)ws_size;
    const float* x  = (const float*)d_in[0];
    const float* gw = (const float*)d_in[1];
    const float* gb = (const float*)d_in[2];
    const float* ew = (const float*)d_in[3];
    const float* eb = (const float*)d_in[4];
    float* out = (float*)d_out;

    char* ws = (char*)d_ws;
    float* xn   = (float*)ws;                              ws += (size_t)BATCH * SEQ * 4;       // 16.78 MB
    float* gemm = (float*)ws;                              ws += (size_t)2 * BATCH * PRED * 4;  // 47.19 MB
    int*   e0a  = (int*)ws;                                ws += (size_t)BATCH * 4;
    int*   e1a  = (int*)ws;                                ws += (size_t)BATCH * 4;
    float* g0a  = (float*)ws;                              ws += (size_t)BATCH * 4;
    float* g1a  = (float*)ws;                              ws += (size_t)BATCH * 4;
    float* mua  = (float*)ws;                              ws += (size_t)BATCH * 4;
    float* sda  = (float*)ws;                              ws += (size_t)BATCH * 4;
    int*   rowlist = (int*)ws;                             ws += (size_t)NEXP * BATCH * 4;      // 0.52 MB
    int*   counts  = (int*)ws;

    moe_stage0_zero<<<1, 32, 0, stream>>>(counts);
    moe_stage1<<<BATCH, 256, 0, stream>>>(x, gw, gb, xn, e0a, e1a, g0a, g1a,
                                          mua, sda, counts, rowlist);
    moe_stage2<<<dim3(6, BATCH / 16, NEXP), 256, 0, stream>>>(xn, ew, counts,
                                                              rowlist, gemm);
    moe_stage3<<<BATCH, 256, 0, stream>>>(gemm, eb, e0a, e1a, g0a, g1a,
                                          mua, sda, out);
}